// CausalFeatureTransformer_52132313039065
// MI455X (gfx1250) — compile-verified
//
#include <hip/hip_runtime.h>
#include <hip/hip_bf16.h>

typedef __attribute__((ext_vector_type(2))) float v2f;
typedef __attribute__((ext_vector_type(8))) float v8f;

#define XS  66   // Xn row stride (floats): conflict-free banks, 8B-aligned rows
#define UTS 66   // U-transposed row stride
#define SS  132  // score row stride (floats): 528B rows -> 8B aligned, conflict-free

__device__ __forceinline__ float block_reduce_sum(float v, float* red, int tid) {
    red[tid] = v;
    __syncthreads();
    #pragma unroll
    for (int s = 64; s > 0; s >>= 1) {
        if (tid < s) red[tid] += red[tid + s];
        __syncthreads();
    }
    float r = red[0];
    __syncthreads();
    return r;
}

__global__ __launch_bounds__(128)
void CausalFeatureTransformer_52132313039065_kernel(
        const float* __restrict__ Z, const float* __restrict__ A_full,
        const float* __restrict__ feat_emb, const float* __restrict__ label_token,
        const float* __restrict__ wq, const float* __restrict__ bq,
        const float* __restrict__ wk, const float* __restrict__ bk,
        const float* __restrict__ wv, const float* __restrict__ bv,
        const float* __restrict__ wo, const float* __restrict__ bo,
        const float* __restrict__ w1, const float* __restrict__ b1,
        const float* __restrict__ w2, const float* __restrict__ b2,
        const float* __restrict__ alphap,
        const float* __restrict__ g1, const float* __restrict__ be1,
        const float* __restrict__ g2, const float* __restrict__ be2,
        float* __restrict__ out)
{
    __shared__ float sXn[144 * XS];       // rows 0..128 live, 129..143 zero pad
    __shared__ float sUt[16 * UTS];       // U transposed: [head][e]; rows 4..15 zero
    __shared__ float sScore[16 * SS];     // rows 0..3 live (cols 0..128); rest zero
    __shared__ float sY[4][64];
    __shared__ float sRed[128];
    __shared__ float sZn[128];
    __shared__ float sMu[128], sVar[128];
    __shared__ float sQlab[64];
    __shared__ float sC[4];
    __shared__ float sAB[132];            // alpha * A_full[j, 128], j = 0..128
    __shared__ float sG1[64], sBe1[64];
    __shared__ float sA64[64], sXrow[64], sT[64];
    __shared__ float sH[128];

    const int tid  = threadIdx.x;
    const int blk  = blockIdx.x;
    const int w    = tid >> 5;        // wave id (0..3)
    const int lane = tid & 31;
    const int lh   = lane >> 4;       // half-wave (0/1)
    const int mr   = lane & 15;

    // ---- Zero the padded WMMA operand regions once (no guards needed later) ----
    for (int i = tid; i < 15 * XS; i += 128) sXn[129 * XS + i] = 0.0f;   // rows 129..143
    for (int i = tid; i < 16 * UTS; i += 128) sUt[i] = 0.0f;
    for (int i = tid; i < 16 * SS;  i += 128) sScore[i] = 0.0f;

    // ---- Stage the attention bias column: sAB[j] = alpha * A_full[j, 128] ----
    const float alpha = alphap[0];
    sAB[tid] = alpha * A_full[tid * 129 + 128];
    if (tid == 0) sAB[128] = alpha * A_full[128 * 129 + 128];

    // ---- Phase 0a: LN(label_token) -> Xn row 128 (sample-independent) ----
    float lt = (tid < 64) ? label_token[tid] : 0.0f;
    float s0 = block_reduce_sum(lt, sRed, tid);
    float lm = s0 * (1.0f / 64.0f);
    float ld = (tid < 64) ? (lt - lm) : 0.0f;
    float lv = block_reduce_sum(ld * ld, sRed, tid) * (1.0f / 64.0f);
    float lr = rsqrtf(lv + 1e-5f);
    if (tid < 64) sXn[128 * XS + tid] = ld * lr * g1[tid] + be1[tid];
    __syncthreads();

    // ---- Phase 0b: Qlab = Xn128 @ wq + bq ----
    if (tid < 64) {
        float acc = bq[tid];
        for (int e = 0; e < 64; e++) acc += sXn[128 * XS + e] * wq[e * 64 + tid];
        sQlab[tid] = acc;
    }
    __syncthreads();

    // ---- Phase 0c: u (transposed), c, feat_emb row stats, preload g1/be1 ----
    if (tid < 64) {
        const int e = tid;
        #pragma unroll
        for (int h = 0; h < 4; h++) {
            float acc = 0.0f;
            for (int d = 0; d < 16; d++) acc += wk[e * 64 + h * 16 + d] * sQlab[h * 16 + d];
            sUt[h * UTS + e] = acc;
        }
        sG1[tid]  = g1[tid];
        sBe1[tid] = be1[tid];
    }
    if (tid < 4) {
        float acc = 0.0f;
        for (int d = 0; d < 16; d++) acc += bk[tid * 16 + d] * sQlab[tid * 16 + d];
        sC[tid] = acc;
    }
    {
        const int j = tid;  // 128 threads, one feat_emb row each
        float s = 0.0f;
        for (int k = 0; k < 64; k++) s += feat_emb[j * 64 + k];
        float mj = s * (1.0f / 64.0f);
        float q = 0.0f;
        for (int k = 0; k < 64; k++) { float d = feat_emb[j * 64 + k] - mj; q += d * d; }
        sMu[j]  = mj;
        sVar[j] = q * (1.0f / 64.0f);
    }
    __syncthreads();

    // ---- Phase 1: Zn = LN(Z row) over 128 features ----
    float z  = Z[(size_t)blk * 128 + tid];
    float zm = block_reduce_sum(z, sRed, tid) * (1.0f / 128.0f);
    float zd = z - zm;
    float zv = block_reduce_sum(zd * zd, sRed, tid) * (1.0f / 128.0f);
    sZn[tid] = zd * rsqrtf(zv + 1e-5f);
    __syncthreads();

    // ---- Phase 2: Xn rows 0..127 via factored LN of (Zn_j * feat_emb[j,:]) ----
    #pragma unroll 4
    for (int i = 0; i < 64; i++) {
        int idx = tid + i * 128;           // == j*64 + k, coalesced feat_emb read
        int j = idx >> 6, k = idx & 63;
        float zn = sZn[j];
        float fe = feat_emb[idx];
        float rj = rsqrtf(zn * zn * sVar[j] + 1e-5f);
        sXn[j * XS + k] = (zn * fe - zn * sMu[j]) * rj * sG1[k] + sBe1[k];
    }
    __syncthreads();

    // ---- Phase 3: scores[h][j] = 0.25*(Xn[j].u_h + c_h) + alpha*A_full[j,128]
    //      WMMA f32 16x16x4: A = Xn tile rows (zero-padded), B = U^T (zero-padded cols)
    for (int mt = w; mt < 9; mt += 4) {
        v8f acc = {0.f, 0.f, 0.f, 0.f, 0.f, 0.f, 0.f, 0.f};
        const int jbase = mt * 16 + mr;          // A row for this lane (<= 143, in bounds)
        #pragma unroll
        for (int kk = 0; kk < 64; kk += 4) {
            const int k0 = kk + 2 * lh;          // this lane's K pair: k0, k0+1
            v2f a, b;
            a[0] = sXn[jbase * XS + k0];
            a[1] = sXn[jbase * XS + k0 + 1];
            b[0] = sUt[mr * UTS + k0];
            b[1] = sUt[mr * UTS + k0 + 1];
            acc = __builtin_amdgcn_wmma_f32_16x16x4_f32(
                false, a, false, b, (short)0, acc, false, false);
        }
        if (mt < 8) {                            // all 16 rows valid: unconditional writeback
            if (mr < 4) {
                const float ch = sC[mr];
                #pragma unroll
                for (int r = 0; r < 8; r++) {
                    const int j = mt * 16 + r + 8 * lh;   // D: M = r + 8*half
                    sScore[mr * SS + j] = 0.25f * (acc[r] + ch) + sAB[j];
                }
            }
        } else {                                 // tile 8: only j == 128 (r=0, half 0) is live
            if (mr < 4 && lh == 0) {
                sScore[mr * SS + 128] = 0.25f * (acc[0] + sC[mr]) + sAB[128];
            }
        }
    }
    __syncthreads();

    // ---- Phase 4: softmax over j (one wave per head); pad cols stay zero ----
    {
        const int h = w;
        float mx = -1e30f;
        for (int j = lane; j < 129; j += 32) mx = fmaxf(mx, sScore[h * SS + j]);
        #pragma unroll
        for (int off = 16; off > 0; off >>= 1) mx = fmaxf(mx, __shfl_xor(mx, off, 32));
        float sum = 0.0f;
        for (int j = lane; j < 129; j += 32) {
            float e = expf(sScore[h * SS + j] - mx);
            sScore[h * SS + j] = e;
            sum += e;
        }
        #pragma unroll
        for (int off = 16; off > 0; off >>= 1) sum += __shfl_xor(sum, off, 32);
        const float inv = 1.0f / sum;
        for (int j = lane; j < 129; j += 32) sScore[h * SS + j] *= inv;
    }
    __syncthreads();

    // ---- Phase 5: y[h,:] = attn[h,:] @ Xn  (M=4 pad 16, K=129 pad 132, N-tile = wave) ----
    {
        v8f acc = {0.f, 0.f, 0.f, 0.f, 0.f, 0.f, 0.f, 0.f};
        const int e = w * 16 + mr;      // B column (N) handled by this lane
        #pragma unroll
        for (int kk = 0; kk < 132; kk += 4) {
            const int k0 = kk + 2 * lh;
            v2f a, b;
            a[0] = sScore[mr * SS + k0];       // rows 4..15 and cols >128 are zero
            a[1] = sScore[mr * SS + k0 + 1];
            b[0] = sXn[k0 * XS + e];           // rows 129..143 are zero
            b[1] = sXn[(k0 + 1) * XS + e];
            acc = __builtin_amdgcn_wmma_f32_16x16x4_f32(
                false, a, false, b, (short)0, acc, false, false);
        }
        if (lh == 0) {                  // heads live in D rows M=0..3 (half 0, r<4)
            #pragma unroll
            for (int r = 0; r < 4; r++) sY[r][e] = acc[r];
        }
    }
    __syncthreads();

    // ---- Phase 6: attn_out = y @ wv + bv ; o = attn_out @ wo + bo ; Xrow = label + o ----
    if (tid < 64) {
        const int hd = tid, h = hd >> 4;
        float acc = bv[hd];
        for (int e = 0; e < 64; e++) acc += sY[h][e] * wv[e * 64 + hd];
        sA64[hd] = acc;
    }
    __syncthreads();
    if (tid < 64) {
        const int k = tid;
        float acc = bo[k];
        for (int hd = 0; hd < 64; hd++) acc += sA64[hd] * wo[hd * 64 + k];
        sXrow[k] = label_token[k] + acc;
    }
    __syncthreads();

    // ---- Phase 7: t = LN(Xrow)*g2 + be2 ----
    float xv = (tid < 64) ? sXrow[tid] : 0.0f;
    float m2 = block_reduce_sum(xv, sRed, tid) * (1.0f / 64.0f);
    float dv = (tid < 64) ? (xv - m2) : 0.0f;
    float v2 = block_reduce_sum(dv * dv, sRed, tid) * (1.0f / 64.0f);
    if (tid < 64) sT[tid] = dv * rsqrtf(v2 + 1e-5f) * g2[tid] + be2[tid];
    __syncthreads();

    // ---- Phase 8: FFN (exact gelu) + residual, emit last-token row ----
    {
        const int p = tid;  // 128 hidden units
        float acc = b1[p];
        for (int k = 0; k < 64; k++) acc += sT[k] * w1[k * 128 + p];
        sH[p] = 0.5f * acc * (1.0f + erff(acc * 0.70710678118654752f));
    }
    __syncthreads();
    if (tid < 64) {
        const int k = tid;
        float acc = b2[k];
        for (int p = 0; p < 128; p++) acc += sH[p] * w2[p * 64 + k];
        out[(size_t)blk * 64 + k] = sXrow[k] + acc;
    }
}

extern "C" void kernel_launch(void* const* d_in, const int* in_sizes, int n_in,
                              void* d_out, int out_size, void* d_ws, size_t ws_size,
                              hipStream_t stream) {
    (void)n_in; (void)d_ws; (void)ws_size; (void)out_size;
    const float* Z          = (const float*)d_in[0];
    const float* A_full     = (const float*)d_in[1];
    const float* feat_emb   = (const float*)d_in[2];
    const float* label_tok  = (const float*)d_in[3];
    const float* wq         = (const float*)d_in[4];
    const float* bq         = (const float*)d_in[5];
    const float* wk         = (const float*)d_in[6];
    const float* bk         = (const float*)d_in[7];
    const float* wv         = (const float*)d_in[8];
    const float* bv         = (const float*)d_in[9];
    const float* wo         = (const float*)d_in[10];
    const float* bo         = (const float*)d_in[11];
    const float* w1         = (const float*)d_in[12];
    const float* b1         = (const float*)d_in[13];
    const float* w2         = (const float*)d_in[14];
    const float* b2         = (const float*)d_in[15];
    const float* alphap     = (const float*)d_in[16];
    const float* g1         = (const float*)d_in[17];
    const float* be1        = (const float*)d_in[18];
    const float* g2         = (const float*)d_in[19];
    const float* be2        = (const float*)d_in[20];
    float* out              = (float*)d_out;

    const int N = in_sizes[0] / 128;   // 4096 samples
    CausalFeatureTransformer_52132313039065_kernel<<<N, 128, 0, stream>>>(
        Z, A_full, feat_emb, label_tok, wq, bq, wk, bk, wv, bv, wo, bo,
        w1, b1, w2, b2, alphap, g1, be1, g2, be2, out);
}